// EdgeEncoder_69123203662146
// MI455X (gfx1250) — compile-verified
//
#include <hip/hip_runtime.h>
#include <math.h>

// Problem constants (match reference)
#define BB      4
#define NN      2048
#define KK      30
#define DIM_E   128
#define FDIM    39      // NUM_PE + NUM_RBF + 7
#define FPAD    40      // padded to multiple of 4 for 16x16x4 f32 WMMA
#define FSTR    41      // LDS row stride (coprime with 64 banks)
#define EPSF    1e-6f
#define FLTMAX  3.402823466e38f

typedef __attribute__((ext_vector_type(2))) float v2f;
typedef __attribute__((ext_vector_type(8))) float v8f;

__device__ __forceinline__ float signf(float x) {
    return (x > 0.f) ? 1.f : ((x < 0.f) ? -1.f : 0.f);
}

// ---------------------------------------------------------------------------
// Kernel 0: per-node orientation frames O[b,i] (9 floats), zero at padded rows.
//   u2 = normalize(X[i]   - X[N-1])
//   u1 = normalize(X[i+1] - X[N-1])
//   n2 = normalize(cross(u2,u1)); o1 = normalize(u2-u1); c = cross(o1,n2)
//   O9 = [o1.x n2.x c.x  o1.y n2.y c.y  o1.z n2.z c.z]   (valid for 1<=i<=N-3)
// ---------------------------------------------------------------------------
__global__ void frames_kernel(const float* __restrict__ X, float* __restrict__ Ofr) {
    int i = blockIdx.x * blockDim.x + threadIdx.x;
    if (i >= BB * NN) return;
    int b = i / NN, loc = i % NN;
    float o[9];
#pragma unroll
    for (int t = 0; t < 9; ++t) o[t] = 0.f;
    if (loc >= 1 && loc <= NN - 3) {
        const float* Xb = X + (size_t)b * NN * 3;
        float lx = Xb[(NN - 1) * 3 + 0], ly = Xb[(NN - 1) * 3 + 1], lz = Xb[(NN - 1) * 3 + 2];
        float ax = Xb[loc * 3 + 0] - lx, ay = Xb[loc * 3 + 1] - ly, az = Xb[loc * 3 + 2] - lz;
        float bx = Xb[(loc + 1) * 3 + 0] - lx, by = Xb[(loc + 1) * 3 + 1] - ly, bz = Xb[(loc + 1) * 3 + 2] - lz;
        float na = 1.f / fmaxf(sqrtf(ax * ax + ay * ay + az * az), 1e-12f);
        ax *= na; ay *= na; az *= na;                       // u2
        float nb = 1.f / fmaxf(sqrtf(bx * bx + by * by + bz * bz), 1e-12f);
        bx *= nb; by *= nb; bz *= nb;                       // u1
        float cx = ay * bz - az * by, cy = az * bx - ax * bz, cz = ax * by - ay * bx;
        float nc = 1.f / fmaxf(sqrtf(cx * cx + cy * cy + cz * cz), 1e-12f);
        cx *= nc; cy *= nc; cz *= nc;                       // n_2
        float ox = ax - bx, oy = ay - by, oz = az - bz;
        float no = 1.f / fmaxf(sqrtf(ox * ox + oy * oy + oz * oz), 1e-12f);
        ox *= no; oy *= no; oz *= no;                       // o_1
        float wx = oy * cz - oz * cy, wy = oz * cx - ox * cz, wz = ox * cy - oy * cx; // cross(o1,n2)
        o[0] = ox; o[1] = cx; o[2] = wx;
        o[3] = oy; o[4] = cy; o[5] = wy;
        o[6] = oz; o[7] = cz; o[8] = wz;
    }
#pragma unroll
    for (int t = 0; t < 9; ++t) Ofr[(size_t)i * 9 + t] = o[t];
}

// ---------------------------------------------------------------------------
// Kernel 1: top-K neighbor selection. One wave32 per (b,n) row.
// Distances (mask-adjusted) staged in LDS, K=30 rounds of wave argmin
// (tie-break to lower index -> matches jax.lax.top_k stability).
// ---------------------------------------------------------------------------
__global__ void __launch_bounds__(32) topk_kernel(const float* __restrict__ X,
                                                  const float* __restrict__ mask,
                                                  float* __restrict__ Dn,
                                                  int* __restrict__ Eidx) {
    __shared__ float sD[NN];
    int row = blockIdx.x;                  // b*NN + n
    int b = row / NN, n = row % NN;
    int lane = threadIdx.x;
    const float* Xb = X + (size_t)b * NN * 3;
    const float* Mb = mask + (size_t)b * NN;
    float xn = Xb[n * 3 + 0], yn = Xb[n * 3 + 1], zn = Xb[n * 3 + 2];
    float mn = Mb[n];

    float lmax = 0.f;
    for (int j = lane; j < NN; j += 32) {
        float dx = xn - Xb[j * 3 + 0];
        float dy = yn - Xb[j * 3 + 1];
        float dz = zn - Xb[j * 3 + 2];
        float m2 = mn * Mb[j];
        float d = m2 * sqrtf(dx * dx + dy * dy + dz * dz + EPSF);
        sD[j] = d;
        lmax = fmaxf(lmax, d);
    }
#pragma unroll
    for (int m = 16; m >= 1; m >>= 1) lmax = fmaxf(lmax, __shfl_xor(lmax, m, 32));
    __syncthreads();
    // D_adjust = D + (1 - mask2) * D_max
    for (int j = lane; j < NN; j += 32) {
        float m2 = mn * Mb[j];
        sD[j] += (1.f - m2) * lmax;
    }
    __syncthreads();

    for (int it = 0; it < KK; ++it) {
        float bv = FLTMAX;
        int bi = 0x7fffffff;
        for (int j = lane; j < NN; j += 32) {
            float v = sD[j];
            if (v < bv) { bv = v; bi = j; }
        }
#pragma unroll
        for (int m = 16; m >= 1; m >>= 1) {
            float ov = __shfl_xor(bv, m, 32);
            int   oi = __shfl_xor(bi, m, 32);
            if (ov < bv || (ov == bv && oi < bi)) { bv = ov; bi = oi; }
        }
        if (lane == 0) {
            Dn[(size_t)row * KK + it]   = bv;
            Eidx[(size_t)row * KK + it] = bi;
            sD[bi] = FLTMAX;
        }
        __syncthreads();
    }
}

// ---------------------------------------------------------------------------
// Kernel 2: per-edge features -> fp32 WMMA GEMM (39x128, padded to 40) -> LN.
// 8 waves/block; each wave owns one 16-edge M-tile (2 tiles per (b,n) row).
// ---------------------------------------------------------------------------
__global__ void __launch_bounds__(256) edges_kernel(const float* __restrict__ X,
                                                    const float* __restrict__ W_e,
                                                    const float* __restrict__ b_e,
                                                    const float* __restrict__ gain,
                                                    const float* __restrict__ bias,
                                                    const float* __restrict__ Ofr,
                                                    const float* __restrict__ Dn,
                                                    const int* __restrict__ Eidx,
                                                    float* __restrict__ outE) {
    __shared__ float sW[FPAD][DIM_E];       // 20 KB
    __shared__ float sF[8][16 * FSTR];      // 21 KB  (per-wave feature tiles)
    __shared__ float sBe[DIM_E], sG[DIM_E], sBi[DIM_E];

    __builtin_prefetch(W_e + threadIdx.x, 0, 0);  // global_prefetch_b8

    // Stage W_e (zero-padded row 39) + vectors
    for (int e = threadIdx.x; e < FPAD * DIM_E; e += 256) {
        int r = e >> 7, c = e & 127;
        sW[r][c] = (r < FDIM) ? W_e[r * DIM_E + c] : 0.f;
    }
    if (threadIdx.x < DIM_E) {
        sBe[threadIdx.x] = b_e[threadIdx.x];
        sG[threadIdx.x]  = gain[threadIdx.x];
        sBi[threadIdx.x] = bias[threadIdx.x];
    }
    __syncthreads();

    int wave = threadIdx.x >> 5;
    int lane = threadIdx.x & 31;
    int tile = blockIdx.x * 8 + wave;       // < BB*NN*2 (grid is exact)
    int row  = tile >> 1;                   // b*NN + n
    int b    = row / NN, n = row % NN;
    int k0   = (tile & 1) * 16;
    const float* Xb = X + (size_t)b * NN * 3;
    float* sFw = sF[wave];

    // ---- per-edge features (lanes 0..15, one edge each) ----
    if (lane < 16) {
        int k = k0 + lane;
        bool valid = (k < KK);
        int   jj = valid ? Eidx[(size_t)row * KK + k] : n;
        float d  = valid ? Dn[(size_t)row * KK + k]   : 0.f;

        float f[FPAD];
        // Positional encoding: freq_p = 10000^(-2p/16) = 10^(-p/2)
        const float freq[8] = {1.f, 0.31622776601683794f, 0.1f, 0.031622776601683794f,
                               0.01f, 0.0031622776601683794f, 0.001f, 0.00031622776601683794f};
        float dd = (float)(jj - n);
#pragma unroll
        for (int p = 0; p < 8; ++p) {
            float ang = dd * freq[p];
            f[p]     = cosf(ang);
            f[8 + p] = sinf(ang);
        }
        // RBF: mu_r = 20*r/15, sigma = 1.25
#pragma unroll
        for (int r = 0; r < 16; ++r) {
            float t = (d - 20.f * (float)r / 15.f) * 0.8f;
            f[16 + r] = expf(-t * t);
        }
        // Orientation features
        float Os[9], On[9];
#pragma unroll
        for (int t = 0; t < 9; ++t) Os[t] = Ofr[(size_t)row * 9 + t];
#pragma unroll
        for (int t = 0; t < 9; ++t) On[t] = Ofr[((size_t)b * NN + jj) * 9 + t];
        float d0 = Xb[jj * 3 + 0] - Xb[n * 3 + 0];
        float d1 = Xb[jj * 3 + 1] - Xb[n * 3 + 1];
        float d2 = Xb[jj * 3 + 2] - Xb[n * 3 + 2];
        float du0 = Os[0] * d0 + Os[1] * d1 + Os[2] * d2;
        float du1 = Os[3] * d0 + Os[4] * d1 + Os[5] * d2;
        float du2 = Os[6] * d0 + Os[7] * d1 + Os[8] * d2;
        float ni = 1.f / fmaxf(sqrtf(du0 * du0 + du1 * du1 + du2 * du2), 1e-12f);
        f[32] = du0 * ni; f[33] = du1 * ni; f[34] = du2 * ni;
        // R[i][l] = sum_j Os[3j+i] * On[3j+l]   (O^T * O_nb)
        float R[3][3];
#pragma unroll
        for (int i = 0; i < 3; ++i)
#pragma unroll
            for (int l = 0; l < 3; ++l)
                R[i][l] = Os[i] * On[l] + Os[3 + i] * On[3 + l] + Os[6 + i] * On[6 + l];
        float Rxx = R[0][0], Ryy = R[1][1], Rzz = R[2][2];
        float m0 = 0.5f * sqrtf(fabsf(1.f + Rxx - Ryy - Rzz));
        float m1 = 0.5f * sqrtf(fabsf(1.f - Rxx + Ryy - Rzz));
        float m2 = 0.5f * sqrtf(fabsf(1.f - Rxx - Ryy + Rzz));
        float q0 = signf(R[2][1] - R[1][2]) * m0;
        float q1 = signf(R[0][2] - R[2][0]) * m1;
        float q2 = signf(R[1][0] - R[0][1]) * m2;
        float w  = 0.5f * sqrtf(fmaxf(0.f, 1.f + Rxx + Ryy + Rzz));
        float qi = 1.f / fmaxf(sqrtf(q0 * q0 + q1 * q1 + q2 * q2 + w * w), 1e-12f);
        f[35] = q0 * qi; f[36] = q1 * qi; f[37] = q2 * qi; f[38] = w * qi;
        f[39] = 0.f;
        if (!valid) {
#pragma unroll
            for (int t = 0; t < FPAD; ++t) f[t] = 0.f;
        }
#pragma unroll
        for (int t = 0; t < FPAD; ++t) sFw[lane * FSTR + t] = f[t];
    }
    __syncthreads();

    // ---- fp32 WMMA GEMM: D(16x128) = F(16x40) * W(40x128) ----
    int half = lane >> 4;     // 0: K pair {0,1}, 1: K pair {2,3}
    int l16  = lane & 15;
    v8f acc[8];
#pragma unroll
    for (int nt = 0; nt < 8; ++nt)
#pragma unroll
        for (int v = 0; v < 8; ++v) acc[nt][v] = 0.f;

#pragma unroll
    for (int kk = 0; kk < FPAD / 4; ++kk) {
        int kbase = kk * 4 + half * 2;
        v2f a;
        a.x = sFw[l16 * FSTR + kbase];
        a.y = sFw[l16 * FSTR + kbase + 1];
#pragma unroll
        for (int nt = 0; nt < 8; ++nt) {
            v2f bm;
            bm.x = sW[kbase][nt * 16 + l16];
            bm.y = sW[kbase + 1][nt * 16 + l16];
            acc[nt] = __builtin_amdgcn_wmma_f32_16x16x4_f32(
                false, a, false, bm, (short)0, acc[nt], false, false);
        }
    }

    // ---- bias + LayerNorm (ddof=1) + store ----
    // D layout: VGPR v, lanes 0-15 -> M=v, N=lane; lanes 16-31 -> M=v+8, N=lane-16
#pragma unroll
    for (int v = 0; v < 8; ++v) {
        float e[8];
        float s = 0.f, ss = 0.f;
#pragma unroll
        for (int nt = 0; nt < 8; ++nt) {
            float x = acc[nt][v] + sBe[nt * 16 + l16];
            e[nt] = x;
            s += x; ss += x * x;
        }
#pragma unroll
        for (int m = 1; m <= 8; m <<= 1) {
            s  += __shfl_xor(s, m, 32);
            ss += __shfl_xor(ss, m, 32);
        }
        float mu  = s * (1.f / 128.f);
        float var = (ss - 128.f * mu * mu) * (1.f / 127.f);
        float sd  = sqrtf(fmaxf(var, 0.f));
        float rin = 1.f / (sd + EPSF);
        int M = v + half * 8;
        int k = k0 + M;
        if (k < KK) {
            size_t base = ((size_t)row * KK + k) * DIM_E;
#pragma unroll
            for (int nt = 0; nt < 8; ++nt) {
                int col = nt * 16 + l16;
                outE[base + col] = sG[col] * (e[nt] - mu) * rin + sBi[col];
            }
        }
    }
}

// ---------------------------------------------------------------------------
extern "C" void kernel_launch(void* const* d_in, const int* in_sizes, int n_in,
                              void* d_out, int out_size, void* d_ws, size_t ws_size,
                              hipStream_t stream) {
    const float* X    = (const float*)d_in[0];
    const float* mask = (const float*)d_in[1];
    const float* W_e  = (const float*)d_in[2];
    const float* b_e  = (const float*)d_in[3];
    const float* gain = (const float*)d_in[4];
    const float* bias = (const float*)d_in[5];

    float* outE = (float*)d_out;
    int*   Eidx = (int*)(outE + (size_t)BB * NN * KK * DIM_E);  // tuple tail: E_idx

    float* wsO = (float*)d_ws;                      // BB*NN*9 floats
    float* wsD = wsO + (size_t)BB * NN * 9;         // BB*NN*KK floats

    frames_kernel<<<(BB * NN + 255) / 256, 256, 0, stream>>>(X, wsO);
    topk_kernel<<<BB * NN, 32, 0, stream>>>(X, mask, wsD, Eidx);
    edges_kernel<<<(BB * NN * 2) / 8, 256, 0, stream>>>(X, W_e, b_e, gain, bias,
                                                        wsO, wsD, Eidx, outE);
}